// CDCCModel_18545668784197
// MI455X (gfx1250) — compile-verified
//
#include <hip/hip_runtime.h>
#include <hip/hip_bf16.h>

#define SEQ   17520
#define FEAT  1752
#define CL    10

typedef __attribute__((ext_vector_type(2))) float v2f;
typedef __attribute__((ext_vector_type(8))) float v8f;

__device__ __forceinline__ float fast_sigmoid(float x) {
    return 1.0f / (1.0f + __expf(-x));
}
__device__ __forceinline__ float fast_tanh(float x) {
    float t = __expf(-2.0f * x);
    return (1.0f - t) / (1.0f + t);
}

// ---------------------------------------------------------------------------
// Phase 1: bi-LSTM (hidden=1) over 64 independent sequences.
// blockIdx.x = direction (0 fwd, 1 bwd); threadIdx.x = column m in [0,32)
// (m<16 -> x_t batch m ; m>=16 -> x_t_aug batch m-16)
// Writes k-major hf/hb: h*[s*32 + m]  (coalesced per-step stores).
// ---------------------------------------------------------------------------
__global__ void lstm_kernel(const float* __restrict__ xt,
                            const float* __restrict__ xaug,
                            const float* __restrict__ Wih_f, const float* __restrict__ Whh_f,
                            const float* __restrict__ b_f,
                            const float* __restrict__ Wih_b, const float* __restrict__ Whh_b,
                            const float* __restrict__ b_b,
                            float* __restrict__ hf, float* __restrict__ hb) {
    const int m   = threadIdx.x;          // 0..31
    const int dir = blockIdx.x;           // 0 = forward, 1 = backward
    const float* x = (m < 16) ? (xt + (size_t)m * SEQ)
                              : (xaug + (size_t)(m - 16) * SEQ);
    const float* Wih = dir ? Wih_b : Wih_f;
    const float* Whh = dir ? Whh_b : Whh_f;
    const float* bb  = dir ? b_b   : b_f;
    const float wi0 = Wih[0], wi1 = Wih[1], wi2 = Wih[2], wi3 = Wih[3];
    const float wh0 = Whh[0], wh1 = Whh[1], wh2 = Whh[2], wh3 = Whh[3];
    const float b0 = bb[0], b1 = bb[1], b2 = bb[2], b3 = bb[3];

    float h = 0.0f, c = 0.0f;
    if (dir == 0) {
        for (int s = 0; s < SEQ; ++s) {
            float xv = x[s];
            float gi = fmaf(xv, wi0, fmaf(h, wh0, b0));
            float gf = fmaf(xv, wi1, fmaf(h, wh1, b1));
            float gg = fmaf(xv, wi2, fmaf(h, wh2, b2));
            float go = fmaf(xv, wi3, fmaf(h, wh3, b3));
            float ig = fast_sigmoid(gi);
            float fg = fast_sigmoid(gf);
            float g  = fast_tanh(gg);
            float og = fast_sigmoid(go);
            c = fmaf(fg, c, ig * g);
            h = og * fast_tanh(c);
            hf[(size_t)s * 32 + m] = h;
        }
    } else {
        for (int s = SEQ - 1; s >= 0; --s) {
            float xv = x[s];
            float gi = fmaf(xv, wi0, fmaf(h, wh0, b0));
            float gf = fmaf(xv, wi1, fmaf(h, wh1, b1));
            float gg = fmaf(xv, wi2, fmaf(h, wh2, b2));
            float go = fmaf(xv, wi3, fmaf(h, wh3, b3));
            float ig = fast_sigmoid(gi);
            float fg = fast_sigmoid(gf);
            float g  = fast_tanh(gg);
            float og = fast_sigmoid(go);
            c = fmaf(fg, c, ig * g);
            h = og * fast_tanh(c);
            hb[(size_t)s * 32 + m] = h;
        }
    }
}

// Combine directions and transpose to m-major: h[m*SEQ + s].
__global__ void combine_kernel(const float* __restrict__ hf,
                               const float* __restrict__ hb,
                               const float* __restrict__ Wl,
                               const float* __restrict__ bl,
                               float* __restrict__ h) {
    int gid = blockIdx.x * blockDim.x + threadIdx.x;
    if (gid < 32 * SEQ) {
        int m = gid / SEQ;
        int s = gid - m * SEQ;
        size_t src = (size_t)s * 32 + m;
        h[gid] = fmaf(hf[src], Wl[0], fmaf(hb[src], Wl[1], bl[0]));
    }
}

// ---------------------------------------------------------------------------
// Phase 2: weight-streaming GEMM with V_WMMA_F32_16X16X4_F32.
//   Z[m][n] = bias[n] + sum_k W[n][k] * H[m][k]     (H, Z m-major [32][dim])
// blockDim = 128 (4 waves). Wave w owns n-tile (blockIdx.x*4 + w), clamped.
// Double-buffered LDS, ONE barrier per 32-k chunk:
//   W tile 16n x 32k per wave (row stride 36), H tile 32m x 32k shared.
// A frag: float2 from Wt;  B frags: single float2 from Ht (no shuffles).
// ---------------------------------------------------------------------------
#define KT 32
#define LSTR 36   // LDS row stride (floats): 16B-aligned rows + bank spread

__global__ void gemm_mm(const float* __restrict__ W, const float* __restrict__ bias,
                        const float* __restrict__ H, float* __restrict__ Z,
                        int N, int K) {
    __shared__ float Wt[2][4][16 * LSTR];
    __shared__ float Ht[2][32 * LSTR];

    const int tid  = threadIdx.x;      // 0..127
    const int wave = tid >> 5;         // 0..3
    const int lane = tid & 31;         // 0..31
    const int l16  = lane & 15;
    const int half = lane >> 4;        // 0 or 1

    const int ntiles = (N + 15) >> 4;
    int tile = blockIdx.x * 4 + wave;
    if (tile >= ntiles) tile = ntiles - 1;      // redundant waves redo last tile
    const int n0 = tile * 16;

    // staging roles
    const int wrow = lane >> 1;                 // W: 0..15 row within tile
    const int wk   = (lane & 1) * 16;           // W: which 16-k half
    int grow = n0 + wrow; if (grow >= N) grow = N - 1;
    const float* wr = W + (size_t)grow * K;
    const int hrow = tid >> 2;                  // H: m row 0..31 (4 threads per row)
    const int hk   = (tid & 3) * 8;             // H: k start within chunk (2 float4)
    const float* hr = H + (size_t)hrow * K;

    float* wdst0 = &Wt[0][wave][wrow * LSTR + wk];
    float* wdst1 = &Wt[1][wave][wrow * LSTR + wk];
    float* hdst0 = &Ht[0][hrow * LSTR + hk];
    float* hdst1 = &Ht[1][hrow * LSTR + hk];

    v8f acc0 = {};   // m = 0..15
    v8f acc1 = {};   // m = 16..31

    const int nch = (K + KT - 1) / KT;
    const float4 zero4 = make_float4(0.0f, 0.0f, 0.0f, 0.0f);

    for (int c = 0; c < nch; ++c) {
        const int k0 = c * KT;
        // ---- load chunk c: issue all global loads, then store to LDS buf ----
        float4 w0, w1, w2, w3, h0, h1;
        if (k0 + KT <= K) {                     // uniform fast path
            const float* wp = wr + k0 + wk;
            w0 = *(const float4*)(wp + 0);
            w1 = *(const float4*)(wp + 4);
            w2 = *(const float4*)(wp + 8);
            w3 = *(const float4*)(wp + 12);
            const float* hp = hr + k0 + hk;
            h0 = *(const float4*)(hp + 0);
            h1 = *(const float4*)(hp + 4);
        } else {                                // tail chunk (at most one)
            int kw = k0 + wk;
            w0 = (kw + 0  < K) ? *(const float4*)(wr + kw + 0)  : zero4;
            w1 = (kw + 4  < K) ? *(const float4*)(wr + kw + 4)  : zero4;
            w2 = (kw + 8  < K) ? *(const float4*)(wr + kw + 8)  : zero4;
            w3 = (kw + 12 < K) ? *(const float4*)(wr + kw + 12) : zero4;
            int kh = k0 + hk;
            h0 = (kh + 0 < K) ? *(const float4*)(hr + kh + 0) : zero4;
            h1 = (kh + 4 < K) ? *(const float4*)(hr + kh + 4) : zero4;
        }
        // warm L2 for next chunk of both streams
        if (k0 + KT + wk < K) __builtin_prefetch(wr + k0 + KT + wk, 0, 0);
        if (k0 + KT + hk < K) __builtin_prefetch(hr + k0 + KT + hk, 0, 0);

        float* wdst = (c & 1) ? wdst1 : wdst0;
        float* hdst = (c & 1) ? hdst1 : hdst0;
        *(float4*)(wdst + 0)  = w0;
        *(float4*)(wdst + 4)  = w1;
        *(float4*)(wdst + 8)  = w2;
        *(float4*)(wdst + 12) = w3;
        *(float4*)(hdst + 0)  = h0;
        *(float4*)(hdst + 4)  = h1;
        __syncthreads();

        // ---- 8 k-steps of 2x v_wmma_f32_16x16x4_f32 ----
        const float* wtb = (c & 1) ? &Wt[1][wave][0] : &Wt[0][wave][0];
        const float* htb = (c & 1) ? &Ht[1][0]       : &Ht[0][0];
#pragma unroll
        for (int kk = 0; kk < KT; kk += 4) {
            const int ksel = kk + half * 2;
            // A 16x4 f32: lanes 0-15 K=kk,kk+1 (M=l16); lanes 16-31 K=kk+2,kk+3
            v2f a  = *(const v2f*)(wtb + l16 * LSTR + ksel);
            // B 4x16 f32 (mirrored): lanes 0-15 K=kk,kk+1 (N=m); lanes 16-31 K=kk+2,kk+3
            v2f bA = *(const v2f*)(htb + l16 * LSTR + ksel);
            v2f bB = *(const v2f*)(htb + (16 + l16) * LSTR + ksel);
            acc0 = __builtin_amdgcn_wmma_f32_16x16x4_f32(false, a, false, bA,
                                                         (short)0, acc0, false, false);
            acc1 = __builtin_amdgcn_wmma_f32_16x16x4_f32(false, a, false, bB,
                                                         (short)0, acc1, false, false);
        }
        // no trailing barrier needed: next chunk writes the other LDS buffer,
        // and the next barrier keeps waves within one chunk of each other.
    }

    // ---- epilogue: each lane holds 8 consecutive n for fixed m ----
    // acc0: m = l16 ; acc1: m = 16+l16 ; n = n0 + 8*half + v
    const int nb = n0 + 8 * half;
    float* z0 = Z + (size_t)l16 * N;
    float* z1 = Z + (size_t)(16 + l16) * N;
    if (nb + 7 < N) {
        float4 bv0 = *(const float4*)(bias + nb);
        float4 bv1 = *(const float4*)(bias + nb + 4);
        float4 o;
        o.x = acc0[0] + bv0.x; o.y = acc0[1] + bv0.y;
        o.z = acc0[2] + bv0.z; o.w = acc0[3] + bv0.w;
        *(float4*)(z0 + nb) = o;
        o.x = acc0[4] + bv1.x; o.y = acc0[5] + bv1.y;
        o.z = acc0[6] + bv1.z; o.w = acc0[7] + bv1.w;
        *(float4*)(z0 + nb + 4) = o;
        o.x = acc1[0] + bv0.x; o.y = acc1[1] + bv0.y;
        o.z = acc1[2] + bv0.z; o.w = acc1[3] + bv0.w;
        *(float4*)(z1 + nb) = o;
        o.x = acc1[4] + bv1.x; o.y = acc1[5] + bv1.y;
        o.z = acc1[6] + bv1.z; o.w = acc1[7] + bv1.w;
        *(float4*)(z1 + nb + 4) = o;
    } else {
#pragma unroll
        for (int v = 0; v < 8; ++v) {
            int n = nb + v;
            if (n < N) {
                float bv = bias[n];
                z0[n] = acc0[v] + bv;
                z1[n] = acc1[v] + bv;
            }
        }
    }
}

// ---------------------------------------------------------------------------
// Phase 3 epilogues (m-major Z)
// ---------------------------------------------------------------------------

// L2-normalize each row m of Z[32][N] over N, write out[m*N + n].
__global__ void inst_norm_kernel(const float* __restrict__ Z, float* __restrict__ out,
                                 int N) {
    __shared__ float red[256];
    const int m = blockIdx.x;            // 0..31
    const float* zr = Z + (size_t)m * N;
    float s = 0.0f;
    for (int n = threadIdx.x; n < N; n += 256) {
        float z = zr[n];
        s = fmaf(z, z, s);
    }
    red[threadIdx.x] = s;
    __syncthreads();
    for (int off = 128; off > 0; off >>= 1) {
        if (threadIdx.x < off) red[threadIdx.x] += red[threadIdx.x + off];
        __syncthreads();
    }
    float inv = 1.0f / fmaxf(sqrtf(red[0]), 1e-12f);
    for (int n = threadIdx.x; n < N; n += 256) {
        out[(size_t)m * N + n] = zr[n] * inv;
    }
}

// Softmax over the 10 logits of each of 32 rows; out[m*10 + n].
__global__ void softmax10_kernel(const float* __restrict__ Z, float* __restrict__ out) {
    const int m = threadIdx.x;           // 0..31
    const float* zr = Z + (size_t)m * CL;
    float v[CL];
    float mx = -3.4e38f;
#pragma unroll
    for (int n = 0; n < CL; ++n) { v[n] = zr[n]; mx = fmaxf(mx, v[n]); }
    float sum = 0.0f;
#pragma unroll
    for (int n = 0; n < CL; ++n) { v[n] = __expf(v[n] - mx); sum += v[n]; }
    float inv = 1.0f / sum;
#pragma unroll
    for (int n = 0; n < CL; ++n) out[(size_t)m * CL + n] = v[n] * inv;
}

// ---------------------------------------------------------------------------
extern "C" void kernel_launch(void* const* d_in, const int* in_sizes, int n_in,
                              void* d_out, int out_size, void* d_ws, size_t ws_size,
                              hipStream_t stream) {
    const float* x_t   = (const float*)d_in[0];
    const float* x_aug = (const float*)d_in[1];
    const float* Wih_f = (const float*)d_in[2];
    const float* Whh_f = (const float*)d_in[3];
    const float* b_f   = (const float*)d_in[4];
    const float* Wih_b = (const float*)d_in[5];
    const float* Whh_b = (const float*)d_in[6];
    const float* b_b   = (const float*)d_in[7];
    const float* Wl    = (const float*)d_in[8];
    const float* bl    = (const float*)d_in[9];
    const float* Wi1   = (const float*)d_in[10];
    const float* bi1   = (const float*)d_in[11];
    const float* Wi2   = (const float*)d_in[12];
    const float* bi2   = (const float*)d_in[13];
    const float* Wc1   = (const float*)d_in[14];
    const float* bc1   = (const float*)d_in[15];
    const float* Wc2   = (const float*)d_in[16];
    const float* bc2   = (const float*)d_in[17];
    const float* Wc3   = (const float*)d_in[18];
    const float* bc3   = (const float*)d_in[19];
    const float* Wc4   = (const float*)d_in[20];
    const float* bc4   = (const float*)d_in[21];
    float* out = (float*)d_out;

    // workspace layout (floats)
    float* ws   = (float*)d_ws;
    float* hf   = ws;                                   // SEQ*32 (k-major)
    float* hb   = hf + (size_t)SEQ * 32;                // SEQ*32 (k-major)
    float* h    = hb + (size_t)SEQ * 32;                // 32*SEQ (m-major)
    float* bufA = h  + (size_t)SEQ * 32;                // 32*13140 max (m-major)
    float* bufB = bufA + (size_t)32 * 13140;            // 32*13140 max (m-major)

    // 1) bi-LSTM encoder (64 serial scans over 2 waves)
    lstm_kernel<<<2, 32, 0, stream>>>(x_t, x_aug, Wih_f, Whh_f, b_f,
                                      Wih_b, Whh_b, b_b, hf, hb);
    // 2) combine directions through Linear(2,1), transpose to m-major
    combine_kernel<<<(SEQ * 32 + 255) / 256, 256, 0, stream>>>(hf, hb, Wl, bl, h);

    // tiles = ceil(N/16), blocks = ceil(tiles/4)
    // 3) instance head: 17520 -> 8760 -> 1752, then L2 normalize
    gemm_mm<<<(548 + 3) / 4, 128, 0, stream>>>(Wi1, bi1, h,    bufA, 8760, 17520);
    gemm_mm<<<(110 + 3) / 4, 128, 0, stream>>>(Wi2, bi2, bufA, bufB, 1752, 8760);
    inst_norm_kernel<<<32, 256, 0, stream>>>(bufB, out, FEAT);

    // 4) cluster head: 17520 -> 13140 -> 8760 -> 4380 -> 10, then softmax
    gemm_mm<<<(822 + 3) / 4, 128, 0, stream>>>(Wc1, bc1, h,    bufA, 13140, 17520);
    gemm_mm<<<(548 + 3) / 4, 128, 0, stream>>>(Wc2, bc2, bufA, bufB, 8760, 13140);
    gemm_mm<<<(274 + 3) / 4, 128, 0, stream>>>(Wc3, bc3, bufB, bufA, 4380, 8760);
    gemm_mm<<<1,             128, 0, stream>>>(Wc4, bc4, bufA, bufB, CL,   4380);
    softmax10_kernel<<<1, 32, 0, stream>>>(bufB, out + (size_t)32 * FEAT);
}